// resmlp_models_31267361915450
// MI455X (gfx1250) — compile-verified
//
#include <hip/hip_runtime.h>
#include <math.h>

// ---------------------------------------------------------------------------
// CDNA5 (gfx1250) implementation of 3x3-local attention block.
// Pipeline (all intermediates live in d_ws, L2-resident on MI455X, 192MB L2):
//   K1: q  = x(73728x96) @ Wq^T  + bq, * (C^-0.5 * f_qr)   [fp32 WMMA, W in LDS]
//   K2: kv = x(73728x96) @ Wkv^T + bkv                     [fp32 WMMA, W in LDS]
//   K3: per (b, head, pixel) 3x3 softmax attention (VALU, float4 loads)
//   K4: out = attn(36864x192) @ Wp^T + bp                  [fp32 WMMA, W in LDS]
// Memory-bound (~57MB HBM, ~7 GFLOP) -> exact fp32 WMMA 16x16x4 is free.
// ---------------------------------------------------------------------------

typedef float v2f __attribute__((ext_vector_type(2)));
typedef float v8f __attribute__((ext_vector_type(8)));

// out = (A * W^T + bias) * scale
// A: (M, KD) row-major.  W: (NT*16, KD) row-major, staged in LDS per block.
// One wave owns one 16-row m-tile and ALL NT column tiles: per k-step it does
// 1 global b64 (A pair, reused NT times) + NT ds_load_b64 + NT WMMAs.
// LDS rows padded by +4 dwords: lane bank stride (KD+4)%64 = 36 (K=96) or
// 4 (K=192) -> 16 distinct banks per half-wave, conflict-free b64 reads.
template <int NT, int KD>
__global__ __launch_bounds__(256) void wmma_gemm_lds(
    const float* __restrict__ A, const float* __restrict__ W,
    const float* __restrict__ bias, float* __restrict__ out,
    int M, const float* __restrict__ scale_ptr, float static_scale)
{
  constexpr int N  = NT * 16;
  constexpr int KP = KD + 4;          // padded LDS row stride (dwords)
  __shared__ float wlds[N * KP];

  // ---- cooperative stage W -> LDS (float4 copies, 16B-aligned both sides) ----
  constexpr int KQ = KD / 4;          // float4s per row
  for (int i = threadIdx.x; i < N * KQ; i += 256) {
    const int n = i / KQ;
    const int c = i - n * KQ;
    *(float4*)(&wlds[n * KP + (c << 2)]) =
        *(const float4*)(W + (size_t)n * KD + (c << 2));
  }
  __syncthreads();

  const int lane = threadIdx.x & 31;
  const int wave = threadIdx.x >> 5;
  const int mt   = blockIdx.x * 8 + wave;
  if (mt * 16 >= M) return;           // wave-uniform: EXEC all-1s for WMMA

  // A operand: lane%16 -> M row; lanes 16-31 carry K pair {2,3} (ISA 7.12.2)
  const int mrow = mt * 16 + (lane & 15);
  const int kk   = (lane >> 4) << 1;  // 0 or 2
  const float2* __restrict__ arow = (const float2*)(A + (size_t)mrow * KD);
  const float*  __restrict__ wb   = &wlds[(lane & 15) * KP + kk];

  v8f acc[NT];
#pragma unroll
  for (int nt = 0; nt < NT; ++nt)
    acc[nt] = (v8f){0.f, 0.f, 0.f, 0.f, 0.f, 0.f, 0.f, 0.f};

#pragma unroll 4
  for (int k0 = 0; k0 < KD; k0 += 4) {
    const float2 av = arow[(k0 + kk) >> 1];      // A[mrow][k0+kk .. +1]
    v2f a; a.x = av.x; a.y = av.y;
#pragma unroll
    for (int nt = 0; nt < NT; ++nt) {
      const float2 wv = *(const float2*)(wb + nt * 16 * KP + k0);  // ds_load_b64
      v2f b; b.x = wv.x; b.y = wv.y;
      // D = A(16x4,f32) x B(4x16,f32) + C : exact fp32 accumulation
      acc[nt] = __builtin_amdgcn_wmma_f32_16x16x4_f32(
          false, a, false, b, (short)0, acc[nt], false, false);
    }
  }

  float s = static_scale;
  if (scale_ptr) s *= *scale_ptr;

  // C/D layout: VGPR i holds row (i + 8*(lane>>4)), col = lane&15
  const int rbase = mt * 16 + ((lane >> 4) << 3);
#pragma unroll
  for (int nt = 0; nt < NT; ++nt) {
    const int ncol = nt * 16 + (lane & 15);
    const float bb = bias[ncol];
    float* __restrict__ orow = out + (size_t)rbase * N + ncol;
#pragma unroll
    for (int i = 0; i < 8; ++i)
      orow[(size_t)i * N] = (acc[nt][i] + bb) * s;
  }
}

// ---------------------------------------------------------------------------
// 3x3 local attention. One thread per (b, head, pixel).
// q  : (B,192,96,96)  fp32, scale*f_qr already applied
// kv : (B,192,96,192) fp32; raw-reshape algebra gives, for head n, pixel (h,w):
//   k[d] = kv[b, 2h+(w>=48), 2*(w%48)+0, n*16+d]   (16 contiguous floats)
//   v[d] = kv[b, 2h+(w>=48), 2*(w%48)+1, n*16+d]
// Zero-padded neighbors: logit = 0 (enters softmax denom), v = 0.
// out: (B*9216, 192) with channel = n*16+d (proj-GEMM-ready layout).
// ---------------------------------------------------------------------------
__global__ __launch_bounds__(256) void local_attn3x3(
    const float* __restrict__ q, const float* __restrict__ kv,
    const float* __restrict__ fkr_p, const float* __restrict__ fvr_p,
    float* __restrict__ out)
{
  const int L = 96 * 96;
  const int idx = blockIdx.x * blockDim.x + threadIdx.x;   // [0, 4*12*L)
  const int b = idx / (12 * L);
  const int r = idx - b * (12 * L);
  const int n = r / L;
  const int l = r - n * L;
  const int h2 = l / 96;
  const int w2 = l - h2 * 96;
  const int n16 = n * 16;

  const float fk = *fkr_p;
  const float fv = *fvr_p;

  // ---- load q (16 contiguous floats, 64B aligned) ----
  const int cq = 2 * h2 + (w2 >= 48 ? 1 : 0);
  const int hq = 2 * (w2 - (w2 >= 48 ? 48 : 0)) + (n16 >= 96 ? 1 : 0);
  const int wq = n16 - (n16 >= 96 ? 96 : 0);
  const float4* __restrict__ qp =
      (const float4*)(q + ((((size_t)b * 192 + cq) * 96 + hq) * 96 + wq));
  const float4 q0 = qp[0], q1 = qp[1], q2 = qp[2], q3 = qp[3];

  // ---- pass 1: logits ----
  float lg[9];
#pragma unroll
  for (int j = 0; j < 9; ++j) {
    const int hh = h2 + (j / 3) - 1;
    const int ww = w2 + (j % 3) - 1;
    float d = 0.0f;
    if ((unsigned)hh < 96u && (unsigned)ww < 96u) {
      const int ck = 2 * hh + (ww >= 48 ? 1 : 0);
      const int hk = 2 * (ww - (ww >= 48 ? 48 : 0));     // s=0 -> k row
      const float4* __restrict__ kp =
          (const float4*)(kv + ((((size_t)b * 192 + ck) * 96 + hk) * 192 + n16));
      const float4 k0 = kp[0], k1 = kp[1], k2 = kp[2], k3 = kp[3];
      d = q0.x * k0.x;
      d = fmaf(q0.y, k0.y, d); d = fmaf(q0.z, k0.z, d); d = fmaf(q0.w, k0.w, d);
      d = fmaf(q1.x, k1.x, d); d = fmaf(q1.y, k1.y, d);
      d = fmaf(q1.z, k1.z, d); d = fmaf(q1.w, k1.w, d);
      d = fmaf(q2.x, k2.x, d); d = fmaf(q2.y, k2.y, d);
      d = fmaf(q2.z, k2.z, d); d = fmaf(q2.w, k2.w, d);
      d = fmaf(q3.x, k3.x, d); d = fmaf(q3.y, k3.y, d);
      d = fmaf(q3.z, k3.z, d); d = fmaf(q3.w, k3.w, d);
      d *= fk;
    }
    lg[j] = d;   // zero-padded neighbor -> logit 0 (matches reference)
  }

  // ---- softmax over 9 ----
  float mx = lg[0];
#pragma unroll
  for (int j = 1; j < 9; ++j) mx = fmaxf(mx, lg[j]);
  float e[9];
  float den = 0.0f;
#pragma unroll
  for (int j = 0; j < 9; ++j) { e[j] = __expf(lg[j] - mx); den += e[j]; }

  // ---- pass 2: weighted sum of v (v rows are L0/L2-hot) ----
  float4 a0 = {0, 0, 0, 0}, a1 = {0, 0, 0, 0}, a2 = {0, 0, 0, 0}, a3 = {0, 0, 0, 0};
#pragma unroll
  for (int j = 0; j < 9; ++j) {
    const int hh = h2 + (j / 3) - 1;
    const int ww = w2 + (j % 3) - 1;
    if ((unsigned)hh < 96u && (unsigned)ww < 96u) {
      const int ck = 2 * hh + (ww >= 48 ? 1 : 0);
      const int hk = 2 * (ww - (ww >= 48 ? 48 : 0)) + 1;  // s=1 -> v row
      const float4* __restrict__ vp =
          (const float4*)(kv + ((((size_t)b * 192 + ck) * 96 + hk) * 192 + n16));
      const float4 v0 = vp[0], v1 = vp[1], v2 = vp[2], v3 = vp[3];
      const float w = e[j];
      a0.x = fmaf(w, v0.x, a0.x); a0.y = fmaf(w, v0.y, a0.y);
      a0.z = fmaf(w, v0.z, a0.z); a0.w = fmaf(w, v0.w, a0.w);
      a1.x = fmaf(w, v1.x, a1.x); a1.y = fmaf(w, v1.y, a1.y);
      a1.z = fmaf(w, v1.z, a1.z); a1.w = fmaf(w, v1.w, a1.w);
      a2.x = fmaf(w, v2.x, a2.x); a2.y = fmaf(w, v2.y, a2.y);
      a2.z = fmaf(w, v2.z, a2.z); a2.w = fmaf(w, v2.w, a2.w);
      a3.x = fmaf(w, v3.x, a3.x); a3.y = fmaf(w, v3.y, a3.y);
      a3.z = fmaf(w, v3.z, a3.z); a3.w = fmaf(w, v3.w, a3.w);
    }
  }
  const float inv = fv / den;
  a0.x *= inv; a0.y *= inv; a0.z *= inv; a0.w *= inv;
  a1.x *= inv; a1.y *= inv; a1.z *= inv; a1.w *= inv;
  a2.x *= inv; a2.y *= inv; a2.z *= inv; a2.w *= inv;
  a3.x *= inv; a3.y *= inv; a3.z *= inv; a3.w *= inv;

  float4* __restrict__ op = (float4*)(out + ((size_t)(b * L + l)) * 192 + n16);
  op[0] = a0; op[1] = a1; op[2] = a2; op[3] = a3;
}

extern "C" void kernel_launch(void* const* d_in, const int* in_sizes, int n_in,
                              void* d_out, int out_size, void* d_ws, size_t ws_size,
                              hipStream_t stream) {
  (void)in_sizes; (void)n_in; (void)out_size; (void)ws_size;
  const float* x   = (const float*)d_in[0];   // (4, 9216, 192)
  const float* Wq  = (const float*)d_in[1];   // (96, 96)
  const float* bq  = (const float*)d_in[2];   // (96,)
  const float* Wkv = (const float*)d_in[3];   // (192, 96)
  const float* bkv = (const float*)d_in[4];   // (192,)
  const float* Wp  = (const float*)d_in[5];   // (192, 192)
  const float* bp  = (const float*)d_in[6];   // (192,)
  const float* fqr = (const float*)d_in[7];
  const float* fkr = (const float*)d_in[8];
  const float* fvr = (const float*)d_in[9];
  float* out = (float*)d_out;

  // Workspace partition (floats): q 73728x96, kv 73728x192, attn 36864x192
  float* ws    = (float*)d_ws;
  float* qbuf  = ws;                         // 7,077,888 f  (28.3 MB)
  float* kvbuf = ws + 7077888;               // 14,155,776 f (56.6 MB)
  float* abuf  = ws + 21233664;              // 7,077,888 f  (28.3 MB)

  const float scale = 1.0f / sqrtf(192.0f);  // C^-0.5; f_qr folded on-device

  // K1: q GEMM  M=73728 N=96  K=96  (8 m-tiles/block -> 576 blocks)
  wmma_gemm_lds<6, 96><<<576, 256, 0, stream>>>(
      x, Wq, bq, qbuf, 73728, fqr, scale);
  // K2: kv GEMM M=73728 N=192 K=96
  wmma_gemm_lds<12, 96><<<576, 256, 0, stream>>>(
      x, Wkv, bkv, kvbuf, 73728, nullptr, 1.0f);
  // K3: attention, one thread per (b, head, pixel): 442368 threads
  local_attn3x3<<<442368 / 256, 256, 0, stream>>>(qbuf, kvbuf, fkr, fvr, abuf);
  // K4: proj GEMM M=36864 N=192 K=192 (2304 m-tiles -> 288 blocks)
  wmma_gemm_lds<12, 192><<<288, 256, 0, stream>>>(
      abuf, Wp, bp, out, 36864, nullptr, 1.0f);
}